// SelfAttention_32676111188382
// MI455X (gfx1250) — compile-verified
//
#include <hip/hip_runtime.h>
#include <hip/hip_bf16.h>

// ---------------------------------------------------------------------------
// GQA decode attention for MI455X (gfx1250, wave32, WMMA, TDM).
// Stages:
//   1) qkv = x @ [Wq|Wk|Wv]       (f32 WMMA 16x16x4 GEMM)
//   2) RoPE on q,k                (elementwise)
//   3) flash-decode attention     (TDM-staged KV, f32 WMMA scores + PV)
//   4) out = ctx @ Wo             (f32 WMMA 16x16x4 GEMM)
// ---------------------------------------------------------------------------

typedef __attribute__((ext_vector_type(2))) float v2f;
typedef __attribute__((ext_vector_type(8))) float v8f;
typedef __attribute__((ext_vector_type(4))) unsigned int v4u;
typedef __attribute__((ext_vector_type(4))) int v4i;
typedef __attribute__((ext_vector_type(8))) int v8i;

#define D_MODEL   4096
#define D_KV      1024
#define HEAD_DIM  128
#define NQH       32
#define NKVH      8
#define NREP      4
#define QKV_LD    6144          // 4096 + 1024 + 1024
#define CACHE_T   4096

#if __has_builtin(__builtin_amdgcn_tensor_load_to_lds)
#define HAVE_TDM 1
#else
#define HAVE_TDM 0
#endif

// ---------------------------------------------------------------------------
// Tensor Data Mover: 2D tile (tile_w x tile_h f32 elements) global -> LDS.
// D# bit layout per CDNA5 ISA 8.3/8.4; groups 2/3 zero (<=2D tensor).
// Rows beyond tensor_h are OOB and zero-filled by the TDM.
// ---------------------------------------------------------------------------
#if HAVE_TDM
__device__ __forceinline__ void tdm_load_2d_f32(
    unsigned lds_addr, const void* gptr, unsigned tile_w, unsigned tile_h,
    unsigned tensor_w, unsigned tensor_h, unsigned row_stride_elems) {
  const unsigned long long ga = (unsigned long long)(uintptr_t)gptr;
  v4u g0;
  g0[0] = 1u;                                               // count=1, user D#
  g0[1] = lds_addr;                                         // LDS byte address
  g0[2] = (unsigned)ga;                                     // global_addr[31:0]
  g0[3] = ((unsigned)(ga >> 32) & 0x01FFFFFFu) | (2u << 30);// addr[56:32] | type=2
  v8i g1;
  g1[0] = (int)(2u << 16);                                  // data_size = 4B
  g1[1] = (int)((tensor_w & 0xFFFFu) << 16);                // tensor_dim0 lo16
  g1[2] = (int)((tensor_w >> 16) | ((tensor_h & 0xFFFFu) << 16)); // dim0 hi / dim1 lo
  g1[3] = (int)((tensor_h >> 16) | ((tile_w & 0xFFFFu) << 16));   // dim1 hi / tile_dim0
  g1[4] = (int)(tile_h & 0xFFFFu);                          // tile_dim1 (tile_dim2=0)
  g1[5] = (int)row_stride_elems;                            // tensor_dim0_stride lo32
  g1[6] = 0;                                                // stride0 hi / stride1 lo
  g1[7] = 0;                                                // stride1 hi
  v4i g2 = {0, 0, 0, 0};
  v4i g3 = {0, 0, 0, 0};
#if __clang_major__ >= 23
  v8i g4 = {0, 0, 0, 0, 0, 0, 0, 0};
  __builtin_amdgcn_tensor_load_to_lds(g0, g1, g2, g3, g4, 0);
#else
  __builtin_amdgcn_tensor_load_to_lds(g0, g1, g2, g3, 0);
#endif
}
#endif

// ---------------------------------------------------------------------------
// Generic C(32 x N) = A(32 x 4096) @ W(4096 x N) using V_WMMA_F32_16X16X4_F32.
// Block: 256 threads = 8 waves. Each wave owns a K-slice of 512 and both
// 16-row M-tiles of one 16-column N-tile; partials reduced through LDS.
// ---------------------------------------------------------------------------
__global__ __launch_bounds__(256) void gemm32_wmma_f32(
    const float* __restrict__ A, const float* __restrict__ W,
    float* __restrict__ C, int N, int ldc) {
  const int n0   = blockIdx.x * 16;
  const int tid  = threadIdx.x;
  const int wave = tid >> 5;
  const int lane = tid & 31;
  const int ml   = lane & 15;              // row (A) / col (B) within tile
  const int koff = (lane < 16) ? 0 : 2;    // K offset for upper half-wave
  const int ncol = n0 + ml;

  const int kbeg = wave * 512;
  const int kend = kbeg + 512;

  v8f acc0 = {};   // M rows 0..15
  v8f acc1 = {};   // M rows 16..31

  for (int k = kbeg; k < kend; k += 4) {
    const int kk = k + koff;
    v2f a0, a1, bf;
    a0[0] = A[(size_t)ml * D_MODEL + kk];
    a0[1] = A[(size_t)ml * D_MODEL + kk + 1];
    a1[0] = A[(size_t)(16 + ml) * D_MODEL + kk];
    a1[1] = A[(size_t)(16 + ml) * D_MODEL + kk + 1];
    bf[0] = W[(size_t)kk * N + ncol];
    bf[1] = W[(size_t)(kk + 1) * N + ncol];
    __builtin_prefetch(&W[(size_t)(kk + 32) * N + ncol], 0, 1);
    acc0 = __builtin_amdgcn_wmma_f32_16x16x4_f32(false, a0, false, bf,
                                                 (short)0, acc0, false, false);
    acc1 = __builtin_amdgcn_wmma_f32_16x16x4_f32(false, a1, false, bf,
                                                 (short)0, acc1, false, false);
  }

  // Cross-wave reduction: red[wave][tile(2)][reg(8)][lane(32)]
  __shared__ float red[8 * 512];
  for (int r = 0; r < 8; ++r) {
    red[wave * 512 + (0 * 8 + r) * 32 + lane] = acc0[r];
    red[wave * 512 + (1 * 8 + r) * 32 + lane] = acc1[r];
  }
  __syncthreads();

  // D layout: VGPR r holds row r (lanes 0-15) / row 8+r (lanes 16-31), col=lane&15
  for (int e = tid; e < 512; e += 256) {
    const int m    = e >> 4;
    const int nl   = e & 15;
    const int tile = m >> 4;
    const int mm   = m & 15;
    const int r    = mm & 7;
    const int half = mm >> 3;
    const int srcL = half * 16 + nl;
    float s = 0.f;
    for (int w = 0; w < 8; ++w) s += red[w * 512 + (tile * 8 + r) * 32 + srcL];
    C[(size_t)m * ldc + n0 + nl] = s;
  }
}

// ---------------------------------------------------------------------------
// RoPE on q (32 heads) and k (8 heads) inside the qkv workspace.
// theta_i = 10000^{-i/8192}  (reference: exps = (2i)/d/d with d=128)
// ---------------------------------------------------------------------------
__global__ void rope_kernel(float* __restrict__ qkv,
                            const int* __restrict__ sp_ptr) {
  const int idx = blockIdx.x * blockDim.x + threadIdx.x;
  if (idx >= 32 * 40 * 64) return;           // 32 b * (32 q + 8 k) heads * 64 pairs
  const int i  = idx & 63;
  const int hi = (idx >> 6) % 40;
  const int b  = idx / (40 * 64);

  const float m     = (float)(*sp_ptr);
  const float theta = __expf(-((float)i) * (9.210340371976184f / 8192.0f));
  const float f     = m * theta;
  float s, c;
  __sincosf(f, &s, &c);

  size_t base;
  if (hi < 32) base = (size_t)b * QKV_LD + (size_t)hi * HEAD_DIM + 2 * i;
  else         base = (size_t)b * QKV_LD + D_MODEL + (size_t)(hi - 32) * HEAD_DIM + 2 * i;

  const float x0 = qkv[base];
  const float x1 = qkv[base + 1];
  qkv[base]     = x0 * c - x1 * s;
  qkv[base + 1] = x0 * s + x1 * c;
}

// ---------------------------------------------------------------------------
// Flash-decode attention. Grid: (kv_head, batch). 256 threads = 8 waves.
// KV chunks of 128 positions are DMA'd into LDS by the Tensor Data Mover
// (wave 0 issues, TENSORcnt-tracked); each wave owns one 16-t tile per chunk
// and runs f32 WMMA scores + online softmax + f32 WMMA PV. The fresh k/v at
// t == start_pos (reference's cache update) is folded in at the final merge,
// so the input caches are never mutated.
// LDS (~146 KB): Klds[128][128], Vlds[128][128], Plds[8][16][4],
//                Om[8][4][128], Ml[8][4], Ll[8][4], Sext[4]
// ---------------------------------------------------------------------------
__global__ __launch_bounds__(256, 1) void attn_decode(
    const float* __restrict__ qkv, const float* __restrict__ cache_k,
    const float* __restrict__ cache_v, float* __restrict__ ctx,
    const int* __restrict__ sp_ptr) {
  extern __shared__ float smem[];
  float* Klds = smem;                       // 16384 floats
  float* Vlds = Klds + 128 * 128;           // 16384
  float* Plds = Vlds + 128 * 128;           // 512
  float* Om   = Plds + 8 * 16 * 4;          // 4096
  float* Ml   = Om + 8 * 4 * 128;           // 32
  float* Ll   = Ml + 8 * 4;                 // 32
  float* Sext = Ll + 8 * 4;                 // 4

  const int h    = blockIdx.x;              // kv head
  const int b    = blockIdx.y;              // batch
  const int tid  = threadIdx.x;
  const int wave = tid >> 5;
  const int lane = tid & 31;
  const int ml16 = lane & 15;
  const int koff = (lane < 16) ? 0 : 2;
  const int sp   = *sp_ptr;                 // 4095

  // Q fragments for the whole head_dim, pre-scaled by 1/sqrt(128).
  // A-frag: lane row = rep (valid < NREP, rest zero-padded), K pair at koff.
  const float scale = 0.08838834764831845f;
  v2f qa[32];
  {
    const int  rep   = ml16;
    const bool valid = rep < NREP;
    const float* qp  = qkv + (size_t)b * QKV_LD + (size_t)(h * NREP + rep) * HEAD_DIM;
    for (int s = 0; s < 32; ++s) {
      const int d = s * 4 + koff;
      qa[s][0] = valid ? qp[d] * scale : 0.f;
      qa[s][1] = valid ? qp[d + 1] * scale : 0.f;
    }
  }

  v8f acc[8];
  for (int i = 0; i < 8; ++i) acc[i] = (v8f){};
  float mrun[4] = {-3e38f, -3e38f, -3e38f, -3e38f};
  float lrun[4] = {0.f, 0.f, 0.f, 0.f};

  const float* kbase = cache_k + (size_t)b * CACHE_T * D_KV + (size_t)h * HEAD_DIM;
  const float* vbase = cache_v + (size_t)b * CACHE_T * D_KV + (size_t)h * HEAD_DIM;
  const float* knew  = qkv + (size_t)b * QKV_LD + D_MODEL + (size_t)h * HEAD_DIM;
  const float* vnew  = qkv + (size_t)b * QKV_LD + D_MODEL + D_KV + (size_t)h * HEAD_DIM;

  // Positions 0 .. sp-1 come from the caches; position sp is handled in the
  // epilogue with the freshly computed k/v.
  for (int cb = 0; cb < sp; cb += 128) {
    __syncthreads();   // previous chunk fully consumed

#if HAVE_TDM
    if (wave == 0) {
      const unsigned rows_left = (unsigned)(CACHE_T - cb);   // physical OOB bound
      tdm_load_2d_f32((unsigned)(uintptr_t)Klds, kbase + (size_t)cb * D_KV,
                      HEAD_DIM, 128, HEAD_DIM, rows_left, D_KV);
      tdm_load_2d_f32((unsigned)(uintptr_t)Vlds, vbase + (size_t)cb * D_KV,
                      HEAD_DIM, 128, HEAD_DIM, rows_left, D_KV);
      __builtin_amdgcn_s_wait_tensorcnt(0);
    }
#else
    for (int f = tid; f < 128 * 32; f += 256) {
      const int tl = f >> 5;
      const int d4 = (f & 31) * 4;
      int tg = cb + tl;
      if (tg >= CACHE_T) tg = CACHE_T - 1;          // clamp (masked later)
      *(float4*)(Klds + tl * HEAD_DIM + d4) =
          *(const float4*)(kbase + (size_t)tg * D_KV + d4);
      *(float4*)(Vlds + tl * HEAD_DIM + d4) =
          *(const float4*)(vbase + (size_t)tg * D_KV + d4);
    }
#endif
    __syncthreads();

    const int t0 = wave * 16;

    // Scores S[rep][t] = q . K_t  via 32 WMMAs over the 128-d inner dim.
    v8f s = (v8f){};
    for (int st = 0; st < 32; ++st) {
      v2f kf;
      const int d = st * 4 + koff;
      kf[0] = Klds[(t0 + ml16) * HEAD_DIM + d];
      kf[1] = Klds[(t0 + ml16) * HEAD_DIM + d + 1];
      s = __builtin_amdgcn_wmma_f32_16x16x4_f32(false, qa[st], false, kf,
                                                (short)0, s, false, false);
    }

    // Online softmax per rep; scores for rep r live in lanes 0-15, reg r.
    const int tg = cb + t0 + ml16;
    float p4[4];
    for (int r = 0; r < NREP; ++r) {
      float v = (lane < 16 && tg < sp) ? s[r] : -3e38f;
      for (int off = 16; off >= 1; off >>= 1) v = fmaxf(v, __shfl_xor(v, off, 32));
      const float mnew  = fmaxf(mrun[r], v);
      const float scOld = __expf(mrun[r] - mnew);
      const float p     = (lane < 16 && tg < sp) ? __expf(s[r] - mnew) : 0.f;
      float sum = p;
      for (int off = 16; off >= 1; off >>= 1) sum += __shfl_xor(sum, off, 32);
      lrun[r] = lrun[r] * scOld + sum;
      mrun[r] = mnew;
      for (int dt = 0; dt < 8; ++dt) acc[dt][r] *= scOld;
      p4[r] = p;
    }
    if (lane < 16)
      *(float4*)(Plds + (wave * 16 + lane) * 4) = make_float4(p4[0], p4[1], p4[2], p4[3]);
    __syncthreads();   // P visible in A-fragment order

    // PV: O[rep][d] += P[rep][t] * V[t][d]  (8 d-tiles x 4 k-steps of WMMA)
    v2f af[4];
    for (int tt = 0; tt < 4; ++tt) {
      const int  trow  = tt * 4 + koff;
      const bool valid = ml16 < NREP;
      af[tt][0] = valid ? Plds[(wave * 16 + trow) * 4 + ml16] : 0.f;
      af[tt][1] = valid ? Plds[(wave * 16 + trow + 1) * 4 + ml16] : 0.f;
    }
    for (int dt = 0; dt < 8; ++dt) {
      for (int tt = 0; tt < 4; ++tt) {
        const int trow = t0 + tt * 4 + koff;
        v2f bf;
        bf[0] = Vlds[trow * HEAD_DIM + dt * 16 + ml16];
        bf[1] = Vlds[(trow + 1) * HEAD_DIM + dt * 16 + ml16];
        acc[dt] = __builtin_amdgcn_wmma_f32_16x16x4_f32(false, af[tt], false, bf,
                                                        (short)0, acc[dt], false, false);
      }
    }
  }

  // ---- Flash-decode merge of the 8 wave partials + the t == sp position ----
  __syncthreads();
  if (lane < 16) {
    for (int dt = 0; dt < 8; ++dt)
      for (int r = 0; r < NREP; ++r)
        Om[(wave * 4 + r) * HEAD_DIM + dt * 16 + lane] = acc[dt][r];
  }
  if (lane == 0) {
    for (int r = 0; r < NREP; ++r) {
      Ml[wave * 4 + r] = mrun[r];
      Ll[wave * 4 + r] = lrun[r];
    }
  }
  // Waves 0..3: score of rep r against the fresh key (t == sp).
  if (wave < NREP) {
    const float* qp = qkv + (size_t)b * QKV_LD + (size_t)(h * NREP + wave) * HEAD_DIM;
    float part = 0.f;
    for (int j = lane; j < HEAD_DIM; j += 32) part += qp[j] * knew[j];
    for (int off = 16; off >= 1; off >>= 1) part += __shfl_xor(part, off, 32);
    if (lane == 0) Sext[wave] = part * scale;
  }
  __syncthreads();

  if (tid < HEAD_DIM) {
    const int d = tid;
    const float vn = vnew[d];
    for (int r = 0; r < NREP; ++r) {
      float M8 = -3e38f;
      for (int w = 0; w < 8; ++w) M8 = fmaxf(M8, Ml[w * 4 + r]);
      float L8 = 0.f, O8 = 0.f;
      for (int w = 0; w < 8; ++w) {
        const float f = __expf(Ml[w * 4 + r] - M8);
        L8 += Ll[w * 4 + r] * f;
        O8 += Om[(w * 4 + r) * HEAD_DIM + d] * f;
      }
      // fold in t == sp
      const float se = Sext[r];
      const float Mf = fmaxf(M8, se);
      const float f8 = __expf(M8 - Mf);
      const float pe = __expf(se - Mf);
      const float L  = L8 * f8 + pe;
      const float O  = O8 * f8 + pe * vn;
      ctx[(size_t)b * D_MODEL + (size_t)(h * NREP + r) * HEAD_DIM + d] = O / L;
    }
  }
}

// ---------------------------------------------------------------------------
extern "C" void kernel_launch(void* const* d_in, const int* in_sizes, int n_in,
                              void* d_out, int out_size, void* d_ws, size_t ws_size,
                              hipStream_t stream) {
  const float* x       = (const float*)d_in[0];
  const float* Wq      = (const float*)d_in[1];
  const float* Wk      = (const float*)d_in[2];
  const float* Wv      = (const float*)d_in[3];
  const float* Wo      = (const float*)d_in[4];
  const float* cache_k = (const float*)d_in[5];
  const float* cache_v = (const float*)d_in[6];
  const int*   sp      = (const int*)d_in[7];

  float* qkv  = (float*)d_ws;                 // 32 x 6144
  float* ctxb = qkv + 32 * QKV_LD;            // 32 x 4096
  float* out  = (float*)d_out;                // 32 x 4096

  // 1) fused-by-launch QKV projection into one 32x6144 workspace matrix
  gemm32_wmma_f32<<<D_MODEL / 16, 256, 0, stream>>>(x, Wq, qkv, D_MODEL, QKV_LD);
  gemm32_wmma_f32<<<D_KV / 16, 256, 0, stream>>>(x, Wk, qkv + D_MODEL, D_KV, QKV_LD);
  gemm32_wmma_f32<<<D_KV / 16, 256, 0, stream>>>(x, Wv, qkv + D_MODEL + D_KV, D_KV, QKV_LD);

  // 2) RoPE on q and k
  rope_kernel<<<(32 * 40 * 64 + 255) / 256, 256, 0, stream>>>(qkv, sp);

  // 3) attention (one workgroup per (kv_head, batch))
  const size_t shmem =
      (size_t)(2 * 128 * 128 + 8 * 16 * 4 + 8 * 4 * 128 + 2 * 8 * 4 + 4) * sizeof(float);
  attn_decode<<<dim3(NKVH, 32), 256, shmem, stream>>>(qkv, cache_k, cache_v, ctxb, sp);

  // 4) output projection
  gemm32_wmma_f32<<<D_MODEL / 16, 256, 0, stream>>>(ctxb, Wo, out, D_MODEL, D_MODEL);
}